// Dwt_CnnMLP_38740605010211
// MI455X (gfx1250) — compile-verified
//
#include <hip/hip_runtime.h>
#include <hip/hip_bf16.h>

typedef float v2f __attribute__((ext_vector_type(2)));
typedef float v8f __attribute__((ext_vector_type(8)));

#define B_BATCH 1024
#define NPIX    (B_BATCH * 64)   // 65536 pixels (b,h,w)
#define NROWS   (B_BATCH * 64)   // rows of the raw reshape (B*H*W, C)
#define NTILES  ((NROWS * 4) / 16)  // 16384 16-row tiles of the (M*4, .) GEMM

// db2 analysis filters, reversed (h = dec[::-1]), matching the reference dwt1d.
__constant__ float H0[4] = {0.48296291314469025f, 0.836516303737469f,
                            0.22414386804185735f, -0.12940952255092145f};
__constant__ float H1[4] = {-0.12940952255092145f, -0.22414386804185735f,
                            0.836516303737469f, -0.48296291314469025f};

// ---------------------------------------------------------------------------
// Stage A: two chained db2 DWTs along the channel axis, fused per (pixel,c2).
// Level 1 (N=360, pad 2/2):  lo1[j]/hi1[j] = sum_k H[k] * x[2j+k-2]
// Level 2 (N=180, pad 2/2):  cda[c2] = sum_k H0[k]*hi1[2c2+k-2]   (lo of hi)
//                            cad[c2] = sum_k H1[k]*lo1[2c2+k-2]   (hi of lo)
// Output layout (B,90,8,8) so multi_dwt's raw reshape is a flat row read.
// ---------------------------------------------------------------------------
__global__ void __launch_bounds__(256) dwt2_kernel(const float* __restrict__ x,
                                                   float* __restrict__ cda,
                                                   float* __restrict__ cad) {
  int tid = blockIdx.x * 256 + threadIdx.x;
  if (tid >= NPIX * 90) return;
  int pix = tid / 90;
  int c2  = tid - pix * 90;
  const float* xp = x + (size_t)pix * 360;  // channels contiguous per pixel

  float cdaV = 0.f, cadV = 0.f;
#pragma unroll
  for (int k = 0; k < 4; ++k) {
    int j = 2 * c2 + k - 2;
    if (j < 0 || j >= 180) continue;  // level-2 input is lo1/hi1[:180], zero-padded
    float lo1 = 0.f, hi1 = 0.f;
#pragma unroll
    for (int k2 = 0; k2 < 4; ++k2) {
      int i = 2 * j + k2 - 2;
      float xv = (i >= 0 && i < 360) ? xp[i] : 0.f;
      lo1 = fmaf(H0[k2], xv, lo1);
      hi1 = fmaf(H1[k2], xv, hi1);
    }
    cdaV = fmaf(H0[k], hi1, cdaV);
    cadV = fmaf(H1[k], lo1, cadV);
  }
  int b = pix >> 6, hw = pix & 63;
  size_t o = (size_t)b * 5760 + (size_t)c2 * 64 + hw;  // (B,90,8,8)
  cda[o] = cdaV;
  cad[o] = cadV;
}

// ---------------------------------------------------------------------------
// Stage B: 5-level Haar recursion per flat row (90 -> 45 -> 23 -> 12 -> 6 -> 3).
// One thread per (row, src in {cd,ca}, level-0 branch). Leaves stored in the
// reference's recursion order: dst[seg*3 + e], global leaf = half*16 + seg.
// ---------------------------------------------------------------------------
__global__ void __launch_bounds__(256) haar_kernel(const float* __restrict__ cda,
                                                   const float* __restrict__ cad,
                                                   float* __restrict__ mdcd,
                                                   float* __restrict__ mdca) {
  int tid = blockIdx.x * 256 + threadIdx.x;
  if (tid >= NROWS * 4) return;
  int half   = tid & 1;         // 0 = level-0 lo branch, 1 = hi branch
  int srcsel = (tid >> 1) & 1;  // 0 = cd (from cda), 1 = ca (from cad)
  int r      = tid >> 2;
  const float R2 = 0.70710678118654752440f;
  const float* src = (srcsel ? cad : cda) + (size_t)r * 90;
  float*       dst = (srcsel ? mdca : mdcd) + (size_t)r * 96 + half * 48;
  float sgn = half ? -1.f : 1.f;

  float a0[45];
#pragma unroll
  for (int j = 0; j < 45; ++j) a0[j] = R2 * (src[2 * j] + sgn * src[2 * j + 1]);

  float bufA[48], bufB[48];
  // L1: 45 -> 23 + 23 (odd: zero-pad last)
#pragma unroll
  for (int j = 0; j < 23; ++j) {
    float x0 = a0[2 * j];
    float x1 = (2 * j + 1 < 45) ? a0[2 * j + 1] : 0.f;
    bufA[j]      = R2 * (x0 + x1);
    bufA[23 + j] = R2 * (x0 - x1);
  }
  // L2: 2 segs of 23 -> 4 segs of 12
#pragma unroll
  for (int s = 0; s < 2; ++s)
#pragma unroll
    for (int j = 0; j < 12; ++j) {
      float x0 = bufA[s * 23 + 2 * j];
      float x1 = (2 * j + 1 < 23) ? bufA[s * 23 + 2 * j + 1] : 0.f;
      bufB[(2 * s) * 12 + j]     = R2 * (x0 + x1);
      bufB[(2 * s + 1) * 12 + j] = R2 * (x0 - x1);
    }
  // L3: 4 segs of 12 -> 8 segs of 6
#pragma unroll
  for (int s = 0; s < 4; ++s)
#pragma unroll
    for (int j = 0; j < 6; ++j) {
      float x0 = bufB[s * 12 + 2 * j];
      float x1 = bufB[s * 12 + 2 * j + 1];
      bufA[(2 * s) * 6 + j]     = R2 * (x0 + x1);
      bufA[(2 * s + 1) * 6 + j] = R2 * (x0 - x1);
    }
  // L4: 8 segs of 6 -> 16 segs of 3  (leaves, recursion order)
#pragma unroll
  for (int s = 0; s < 8; ++s)
#pragma unroll
    for (int j = 0; j < 3; ++j) {
      float x0 = bufA[s * 6 + 2 * j];
      float x1 = bufA[s * 6 + 2 * j + 1];
      bufB[(2 * s) * 3 + j]     = R2 * (x0 + x1);
      bufB[(2 * s + 1) * 3 + j] = R2 * (x0 - x1);
    }
#pragma unroll
  for (int i = 0; i < 12; ++i)
    ((float4*)dst)[i] = make_float4(bufB[4 * i], bufB[4 * i + 1],
                                    bufB[4 * i + 2], bufB[4 * i + 3]);
}

// ---------------------------------------------------------------------------
// Reproduce the reference's raw reshapes:
//  z[m2,l,c] -> cd_concat.flat[m2*96+l*3+c] -> leaf_{l2}[r,e2] -> md*[r*96+l2*3+e2]
// ---------------------------------------------------------------------------
__device__ __forceinline__ float z_gather(const float* __restrict__ mdcd,
                                          const float* __restrict__ mdca,
                                          int m2, int l, int c) {
  const float* src = mdcd;
  int ll = l;
  if (l >= 32) { src = mdca; ll = l - 32; }
  int lin  = m2 * 96 + ll * 3 + c;
  int b    = lin / 6144;
  int rem  = lin - b * 6144;
  int ci   = rem >> 6;            // channel in (B,96,8,8)
  int hw   = rem & 63;
  int l2   = ci / 3;              // leaf index 0..31
  int e    = ci - l2 * 3;
  int lin2 = e * 64 + hw;         // leaf reshape (B,3,8,8) -> (B*64,3)
  int rl   = lin2 / 3;
  int e2   = lin2 - rl * 3;
  int r    = b * 64 + rl;
  return src[(size_t)r * 96 + l2 * 3 + e2];
}

// f32 WMMA: D(16x16) = A(16x4) * B(4x16) + C, chained over K.
static __device__ __forceinline__ v8f wmma4(v2f a, v2f b, v8f c) {
  return __builtin_amdgcn_wmma_f32_16x16x4_f32(false, a, false, b, (short)0, c,
                                               false, false);
}

// ---------------------------------------------------------------------------
// Stage C: gather + 3->4 conv + GEMM chain 64->32->16->8 with ReLU.
// 8 waves/block; each wave owns one 16-row tile and a private LDS tile.
// A layout (16x4 f32): lanes 0-15 hold (M=lane, K=0/1), lanes 16-31 (M, K=2/3).
// B layout mirrored: lane&15 = N; lanes<16 hold K=0/1, lanes>=16 hold K=2/3.
// C layout: VGPR j holds (M = j + 8*(lane>>4), N = lane&15).
// ---------------------------------------------------------------------------
__global__ void __launch_bounds__(256) mlp_kernel(const float* __restrict__ mdcd,
                                                  const float* __restrict__ mdca,
                                                  const float* __restrict__ conv_w,
                                                  const float* __restrict__ conv_b,
                                                  const float* __restrict__ w1,
                                                  const float* __restrict__ b1,
                                                  const float* __restrict__ w2,
                                                  const float* __restrict__ b2,
                                                  const float* __restrict__ w3,
                                                  const float* __restrict__ b3,
                                                  float* __restrict__ out) {
  __shared__ float s_w1[32 * 64];
  __shared__ float s_w2[16 * 32];
  __shared__ float s_w3[8 * 16];
  __shared__ float s_b1[32], s_b2[16], s_b3[8], s_cw[12], s_cb[4];
  __shared__ float s_t[8][16 * 64];  // per-wave activation tile (reused per layer)

  for (int i = threadIdx.x; i < 32 * 64; i += 256) s_w1[i] = w1[i];
  for (int i = threadIdx.x; i < 16 * 32; i += 256) s_w2[i] = w2[i];
  if (threadIdx.x < 8 * 16) s_w3[threadIdx.x] = w3[threadIdx.x];
  if (threadIdx.x < 32) s_b1[threadIdx.x] = b1[threadIdx.x];
  if (threadIdx.x < 16) s_b2[threadIdx.x] = b2[threadIdx.x];
  if (threadIdx.x < 8)  s_b3[threadIdx.x] = b3[threadIdx.x];
  if (threadIdx.x < 12) s_cw[threadIdx.x] = conv_w[threadIdx.x];
  if (threadIdx.x < 4)  s_cb[threadIdx.x] = conv_b[threadIdx.x];
  __syncthreads();

  const int w    = threadIdx.x >> 5;
  const int lane = threadIdx.x & 31;
  const int T    = blockIdx.x * 8 + w;
  const int mrow = lane & 15;  // M row for A, N col for B
  const int kp   = lane >> 4;  // K-pair selector / M-group for C

  // ---- gather + pointwise conv: fill A-tile (16 rows x 64 features) ----
  {
    int trow = T * 16 + mrow;
    int m2 = trow >> 2, o = trow & 3;
    float cb  = s_cb[o];
    float cw0 = s_cw[o * 3], cw1 = s_cw[o * 3 + 1], cw2 = s_cw[o * 3 + 2];
    int lbase = kp * 32;  // two lanes per row split the 64 features
    for (int li = 0; li < 32; ++li) {
      int l = lbase + li;
      float v = cb;
      v = fmaf(cw0, z_gather(mdcd, mdca, m2, l, 0), v);
      v = fmaf(cw1, z_gather(mdcd, mdca, m2, l, 1), v);
      v = fmaf(cw2, z_gather(mdcd, mdca, m2, l, 2), v);
      s_t[w][mrow * 64 + l] = fmaxf(v, 0.f);
    }
  }
  __builtin_amdgcn_wave_barrier();

  // ---- layer 1: (16x64) @ w1^T(64x32): 16 K-chunks x 2 N-tiles ----
  v8f acc0 = {};
  v8f acc1 = {};
#pragma unroll
  for (int kk = 0; kk < 16; ++kk) {
    int k0 = kk * 4 + 2 * kp;
    v2f a;  a[0] = s_t[w][mrow * 64 + k0];           a[1] = s_t[w][mrow * 64 + k0 + 1];
    v2f b0; b0[0] = s_w1[mrow * 64 + k0];            b0[1] = s_w1[mrow * 64 + k0 + 1];
    v2f b1v; b1v[0] = s_w1[(16 + mrow) * 64 + k0];   b1v[1] = s_w1[(16 + mrow) * 64 + k0 + 1];
    acc0 = wmma4(a, b0, acc0);
    acc1 = wmma4(a, b1v, acc1);
  }
  __builtin_amdgcn_wave_barrier();
#pragma unroll
  for (int j = 0; j < 8; ++j) {
    int row = j + 8 * kp;
    s_t[w][row * 32 + mrow]      = fmaxf(acc0[j] + s_b1[mrow], 0.f);
    s_t[w][row * 32 + 16 + mrow] = fmaxf(acc1[j] + s_b1[16 + mrow], 0.f);
  }
  __builtin_amdgcn_wave_barrier();

  // ---- layer 2: (16x32) @ w2^T(32x16) ----
  v8f acc2 = {};
#pragma unroll
  for (int kk = 0; kk < 8; ++kk) {
    int k0 = kk * 4 + 2 * kp;
    v2f a;  a[0] = s_t[w][mrow * 32 + k0];  a[1] = s_t[w][mrow * 32 + k0 + 1];
    v2f bv; bv[0] = s_w2[mrow * 32 + k0];   bv[1] = s_w2[mrow * 32 + k0 + 1];
    acc2 = wmma4(a, bv, acc2);
  }
  __builtin_amdgcn_wave_barrier();
#pragma unroll
  for (int j = 0; j < 8; ++j) {
    int row = j + 8 * kp;
    s_t[w][row * 16 + mrow] = fmaxf(acc2[j] + s_b2[mrow], 0.f);
  }
  __builtin_amdgcn_wave_barrier();

  // ---- layer 3: (16x16) @ w3^T(16x8, N zero-padded to 16) ----
  v8f acc3 = {};
#pragma unroll
  for (int kk = 0; kk < 4; ++kk) {
    int k0 = kk * 4 + 2 * kp;
    v2f a;  a[0] = s_t[w][mrow * 16 + k0];  a[1] = s_t[w][mrow * 16 + k0 + 1];
    v2f bv;
    bv[0] = (mrow < 8) ? s_w3[mrow * 16 + k0] : 0.f;
    bv[1] = (mrow < 8) ? s_w3[mrow * 16 + k0 + 1] : 0.f;
    acc3 = wmma4(a, bv, acc3);
  }
  if (mrow < 8) {
    float bb = s_b3[mrow];
#pragma unroll
    for (int j = 0; j < 8; ++j) {
      int row = T * 16 + j + 8 * kp;  // output row = m2*4 + o
      out[(size_t)row * 8 + mrow] = fmaxf(acc3[j] + bb, 0.f);
    }
  }
}

// ---------------------------------------------------------------------------
extern "C" void kernel_launch(void* const* d_in, const int* in_sizes, int n_in,
                              void* d_out, int out_size, void* d_ws, size_t ws_size,
                              hipStream_t stream) {
  (void)in_sizes; (void)n_in; (void)out_size; (void)ws_size;
  const float* x      = (const float*)d_in[0];
  const float* conv_w = (const float*)d_in[1];
  const float* conv_b = (const float*)d_in[2];
  const float* w1     = (const float*)d_in[3];
  const float* b1     = (const float*)d_in[4];
  const float* w2     = (const float*)d_in[5];
  const float* b2     = (const float*)d_in[6];
  const float* w3     = (const float*)d_in[7];
  const float* b3     = (const float*)d_in[8];
  float* out = (float*)d_out;

  float* ws   = (float*)d_ws;
  float* cda  = ws;                                   // (B,90,8,8)
  float* cad  = cda + (size_t)B_BATCH * 5760;         // (B,90,8,8)
  float* mdcd = cad + (size_t)B_BATCH * 5760;         // (NROWS, 32, 3)
  float* mdca = mdcd + (size_t)NROWS * 96;            // (NROWS, 32, 3)

  dwt2_kernel<<<(NPIX * 90 + 255) / 256, 256, 0, stream>>>(x, cda, cad);
  haar_kernel<<<(NROWS * 4 + 255) / 256, 256, 0, stream>>>(cda, cad, mdcd, mdca);
  mlp_kernel<<<NTILES / 8, 256, 0, stream>>>(mdcd, mdca, conv_w, conv_b,
                                             w1, b1, w2, b2, w3, b3, out);
}